// QuantConv_87050397156025
// MI455X (gfx1250) — compile-verified
//
#include <hip/hip_runtime.h>
#include <hip/hip_bf16.h>

// ---------------------------------------------------------------------------
// QuantConv forward for MI455X (gfx1250).
//
// Per-chunk (K=4) partial dot products -> V_WMMA_F32_16X16X4_F32 (exact shape
// match). prob_table (244 KB) lives in WGP LDS with rows 0..7 zeroed so the
// sign(Q)==0 case needs no predication. B weights are stored as 1000.0f so
// the WMMA result IS the table row offset (p*1000, exact in f32) -> no
// v_mul_lo_u32 in the hot loop. rand_idx (151 MB = bandwidth floor ~6.5us
// @ 23.3 TB/s) and the A/B tiles stream through one unified 4-chunk-deep
// register pipeline (~57 outstanding loads/wave, under the 63 LOADcnt cap),
// so no consumption point ever drains the pipeline.
// ---------------------------------------------------------------------------

typedef float v2f __attribute__((ext_vector_type(2)));
typedef float v4f __attribute__((ext_vector_type(4)));
typedef float v8f __attribute__((ext_vector_type(8)));

#define NTILES_N 512      // 8192 / 16
#define NTILES_O 4        // 64 / 16
#define NCH      36       // 144 / 4
#define TABLE_N  61000    // 61 * 1000 dwords
#define ZERO_N   8000     // rows 0..7 -> zero (sign(Q)==0 there)
#define WSCALE   1000.0f  // B weight magnitude: WMMA output = row*1000

// -------------------------- Phase 0a: im2col + quantize into WMMA-A layout --
// A[((nt*36 + c)*32 + lane)*2 + u]:
//   lane 0-15  -> M = lane,     K = 4c + u       (u = 0,1)
//   lane 16-31 -> M = lane-16,  K = 4c + 2 + u
__global__ __launch_bounds__(256) void qconv_prepA(
    const float* __restrict__ x,
    const float* __restrict__ in_min_p,
    const float* __restrict__ in_max_p,
    float* __restrict__ A)
{
    int gid = blockIdx.x * blockDim.x + threadIdx.x;   // 512*36*32 = 589824
    if (gid >= NTILES_N * NCH * 32) return;
    int lane = gid & 31;
    int c    = (gid >> 5) % NCH;
    int t    = gid / (NCH * 32);

    int n  = t * 16 + (lane & 15);       // 0..8191
    int b  = n >> 12;
    int l  = n & 4095;
    int ho = l >> 6, wo = l & 63;
    int k0 = 4 * c + ((lane >> 4) << 1); // 0..142

    const float inmin = *in_min_p;
    const float scale = *in_max_p - inmin;

    float q[2];
#pragma unroll
    for (int u = 0; u < 2; ++u) {
        int k  = k0 + u;                 // ckk = ch*9 + kh*3 + kw
        int ch = k / 9;
        int r  = k - ch * 9;
        int kh = r / 3;
        int kw = r - kh * 3;
        float xv = x[(((b * 16 + ch) * 66) + ho + kh) * 66 + (wo + kw)];
        float v  = ((xv - inmin) / scale) * 15.0f;
        v = rintf(v);                    // round-half-even, matches jnp.round
        q[u] = fminf(fmaxf(v, 0.0f), 15.0f);
    }
    A[gid * 2 + 0] = q[0];
    A[gid * 2 + 1] = q[1];
}

// ---------- Phase 0b: ternary weight split into WMMA-B layout, scaled x1000 -
__global__ __launch_bounds__(256) void qconv_prepB(
    const float* __restrict__ weight,
    float* __restrict__ Bp, float* __restrict__ Bm)
{
    int gid = blockIdx.x * blockDim.x + threadIdx.x;   // 4*36*32 = 4608
    if (gid >= NTILES_O * NCH * 32) return;
    int lane = gid & 31;
    int c    = (gid >> 5) % NCH;
    int ot   = gid / (NCH * 32);
    int o    = ot * 16 + (lane & 15);
    int k0   = 4 * c + ((lane >> 4) << 1);
#pragma unroll
    for (int u = 0; u < 2; ++u) {
        float w = weight[o * 144 + k0 + u];
        Bp[gid * 2 + u] = (w > 0.0f) ? WSCALE : 0.0f;
        Bm[gid * 2 + u] = (w < 0.0f) ? WSCALE : 0.0f;
    }
}

// -------------------------- Phase 0c: per-output bias -----------------------
__global__ void qconv_bias(const float* __restrict__ weight,
                           const float* __restrict__ conv_bias,
                           const float* __restrict__ in_min_p,
                           float* __restrict__ bias)
{
    int o = blockIdx.x * blockDim.x + threadIdx.x;
    if (o >= 64) return;
    float s = 0.0f;
    for (int k = 0; k < 144; ++k) {
        float w = weight[o * 144 + k];
        s += (w > 0.0f) ? 1.0f : ((w < 0.0f) ? -1.0f : 0.0f);
    }
    bias[o] = (*in_min_p) * s + conv_bias[o];
}

// -------------------------- Main: WMMA + LDS-table gather -------------------
__global__ __launch_bounds__(256) void qconv_main(
    const float* __restrict__ A,
    const float* __restrict__ Bp,
    const float* __restrict__ Bm,
    const float* __restrict__ bias,
    const float* __restrict__ table,
    const int*   __restrict__ rand_idx,
    const float* __restrict__ in_min_p,
    const float* __restrict__ in_max_p,
    float* __restrict__ out)
{
    extern __shared__ float lds_table[];               // 244000 B
    const int tid = threadIdx.x;
    {
        v4f* tl = (v4f*)lds_table;
        const v4f* ts = (const v4f*)(table + ZERO_N);
        v4f zv = {0.0f, 0.0f, 0.0f, 0.0f};
        for (int i = tid; i < ZERO_N / 4; i += 256) tl[i] = zv;
        // Copy rows 8..60 with 16 b128 loads in flight per pass (4 passes).
        const int NV = (TABLE_N - ZERO_N) / 4;         // 13250
        for (int base = 0; base < NV; base += 4096) {
            v4f t[16];
#pragma unroll
            for (int k = 0; k < 16; ++k) {
                int idx = base + k * 256 + tid;
                if (idx < NV) t[k] = ts[idx];
            }
#pragma unroll
            for (int k = 0; k < 16; ++k) {
                int idx = base + k * 256 + tid;
                if (idx < NV) tl[ZERO_N / 4 + idx] = t[k];
            }
        }
    }
    __syncthreads();

    const int lane = tid & 31;
    const int wave = tid >> 5;
    const int tile = blockIdx.x * 8 + wave;            // 0..2047
    const int nt   = tile >> 2;                        // n-tile 0..511
    const int ot   = tile & 3;                         // o-tile 0..3

    const int oc    = lane & 15;                       // N within tile
    const int hi    = lane >> 4;                       // M rows 0-7 / 8-15
    const int o     = ot * 16 + oc;
    const int nbase = nt * 16;

    // rand_idx flat index (c, s, n, o): c stride 1048576, s stride 524288,
    // row (j) stride 64. Each instruction: 2 x 64B coalesced segments.
    const int* rbase = rand_idx + (nbase + hi * 8) * 64 + o;

    const float* Ab  = A  + nt * (NCH * 64) + lane * 2;
    const float* Bpb = Bp + ot * (NCH * 64) + lane * 2;
    const float* Bmb = Bm + ot * (NCH * 64) + lane * 2;

    float acc[8];
#pragma unroll
    for (int j = 0; j < 8; ++j) acc[j] = 0.0f;

    // Unified 4-deep prefetch pipeline: rand_idx AND the A/B tiles, so no
    // consumption ever waits on freshly issued loads (no pipeline drain).
    int rbuf[4][16];
    v2f abuf[4], bpbuf[4], bmbuf[4];

#pragma unroll
    for (int u = 0; u < 4; ++u) {
        abuf[u]  = *(const v2f*)(Ab  + u * 64);
        bpbuf[u] = *(const v2f*)(Bpb + u * 64);
        bmbuf[u] = *(const v2f*)(Bmb + u * 64);
        const int* rp = rbase + u * 1048576;
#pragma unroll
        for (int j = 0; j < 8; ++j) {
            rbuf[u][j]     = rp[j * 64];
            rbuf[u][j + 8] = rp[j * 64 + 524288];
        }
    }

    for (int cc = 0; cc < NCH; cc += 4) {
#pragma unroll
        for (int u = 0; u < 4; ++u) {
            const int c = cc + u;
            v8f z = {};
            // D = A(16x4) x B(4x16); B holds 1000.0 -> D = 1000*row, exact.
            v8f pmat = __builtin_amdgcn_wmma_f32_16x16x4_f32(
                false, abuf[u], false, bpbuf[u], (short)0, z, false, false);
            v8f mmat = __builtin_amdgcn_wmma_f32_16x16x4_f32(
                false, abuf[u], false, bmbuf[u], (short)0, z, false, false);

#pragma unroll
            for (int j = 0; j < 8; ++j) {
                int pidx = (int)pmat[j];               // = 1000*p, exact
                int midx = (int)mmat[j];
                float tp = lds_table[pidx + rbuf[u][j]];
                float tm = lds_table[midx + rbuf[u][j + 8]];
                acc[j] += tp - tm;                     // rows<8 are zero
            }

            const int cn = c + 4;
            if (cn < NCH) {                            // refill slot u
                abuf[u]  = *(const v2f*)(Ab  + cn * 64);
                bpbuf[u] = *(const v2f*)(Bpb + cn * 64);
                bmbuf[u] = *(const v2f*)(Bmb + cn * 64);
                const int* rp = rbase + cn * 1048576;
#pragma unroll
                for (int j = 0; j < 8; ++j) {
                    rbuf[u][j]     = rp[j * 64];
                    rbuf[u][j + 8] = rp[j * 64 + 524288];
                }
            }
        }
    }

    const float inmin = *in_min_p;
    const float scale = *in_max_p - inmin;
    const float bo    = bias[o];

    const int b     = nbase >> 12;
    const int lbase = nbase & 4095;
    float* ob = out + (size_t)(b * 64 + o) * 4096 + lbase;
#pragma unroll
    for (int j = 0; j < 8; ++j)
        ob[hi * 8 + j] = acc[j] * scale + bo;          // 2MB, L2-absorbed
}

// ---------------------------------------------------------------------------
extern "C" void kernel_launch(void* const* d_in, const int* in_sizes, int n_in,
                              void* d_out, int out_size, void* d_ws, size_t ws_size,
                              hipStream_t stream) {
    const float* x         = (const float*)d_in[0];
    const float* weight    = (const float*)d_in[1];
    const float* conv_bias = (const float*)d_in[2];
    const float* in_min    = (const float*)d_in[3];
    const float* in_max    = (const float*)d_in[4];
    const float* prob_tab  = (const float*)d_in[5];
    const int*   rand_idx  = (const int*)d_in[6];
    float* out = (float*)d_out;

    float* A    = (float*)d_ws;                        // 4.72 MB
    float* Bp   = A  + NTILES_N * NCH * 64;
    float* Bm   = Bp + NTILES_O * NCH * 64;
    float* bias = Bm + NTILES_O * NCH * 64;

    qconv_prepA<<<(NTILES_N * NCH * 32 + 255) / 256, 256, 0, stream>>>(
        x, in_min, in_max, A);
    qconv_prepB<<<(NTILES_O * NCH * 32 + 255) / 256, 256, 0, stream>>>(
        weight, Bp, Bm);
    qconv_bias<<<1, 64, 0, stream>>>(weight, conv_bias, in_min, bias);

    size_t lds_bytes = TABLE_N * sizeof(float);        // 244000 B
    (void)hipFuncSetAttribute((const void*)qconv_main,
                              hipFuncAttributeMaxDynamicSharedMemorySize,
                              (int)lds_bytes);
    qconv_main<<<256, 256, lds_bytes, stream>>>(
        A, Bp, Bm, bias, prob_tab, rand_idx, in_min, in_max, out);
}